// MMCE_23184233464519
// MI455X (gfx1250) — compile-verified
//
#include <hip/hip_runtime.h>
#include <hip/hip_bf16.h>
#include <math.h>

// MMCE + mean cross-entropy for [N=4096, C=50257] f32 logits.
// Phase 1 is HBM-bound (823 MB @ 23.3 TB/s ~ 35us): single-pass online softmax
// with non-temporal b128 streaming loads (working set 803MB >> 192MB L2).
// Phase 2 (N^2 Laplacian-kernel sums) uses v_wmma_f32_16x16x4_f32 for the
// row-weight contraction over each 16x16 kernel tile.

typedef __attribute__((ext_vector_type(2))) float v2f;
typedef __attribute__((ext_vector_type(4))) float v4f;
typedef __attribute__((ext_vector_type(8))) float v8f;

#define EPSF 1e-12f

// ---------------- online softmax helpers ----------------
__device__ __forceinline__ void online_upd(float x, int c, float& m, float& s, int& idx) {
    if (x > m)       { s = s * expf(m - x) + 1.0f; m = x; idx = c; }
    else if (x == m) { s += 1.0f; if (c < idx) idx = c; }     // first-index tie rule
    else             { s += expf(x - m); }
}
__device__ __forceinline__ void online_merge(float& m, float& s, int& idx,
                                             float m2, float s2, int i2) {
    if (m2 > m)       { s = s * expf(m - m2) + s2; m = m2; idx = i2; }
    else if (m2 == m) { s += s2; if (i2 < idx) idx = i2; }
    else              { s += s2 * expf(m2 - m); }
}

// ---------------- Kernel 1: per-row stats (one block per row) ----------------
__global__ __launch_bounds__(256) void mmce_rowstats(
    const float* __restrict__ input, const int* __restrict__ target, int C,
    float* __restrict__ probs, float* __restrict__ vf, float* __restrict__ vc,
    float* __restrict__ ce, float* __restrict__ mc)
{
    __shared__ float sm[256];
    __shared__ float ss[256];
    __shared__ int   si[256];

    const int row = blockIdx.x;
    const int tid = threadIdx.x;
    const size_t base = (size_t)row * (size_t)C;
    const float* __restrict__ rp = input + base;

    float m = -INFINITY;
    float s = 0.0f;
    int   idx = 0x7fffffff;

    // Align to 16B for b128 streaming loads (C=50257 is not a multiple of 4).
    int pre = (int)((4u - ((unsigned)(base & 3u))) & 3u);
    if (pre > C) pre = C;
    const int nv = (C - pre) >> 2;           // vec4 count
    const int tail0 = pre + (nv << 2);

    for (int c = tid; c < pre; c += 256)
        online_upd(__builtin_nontemporal_load(rp + c), c, m, s, idx);

    const v4f* __restrict__ rp4 = (const v4f*)(rp + pre);
    for (int i = tid; i < nv; i += 256) {
        v4f x = __builtin_nontemporal_load(rp4 + i);
        const int c = pre + (i << 2);
        online_upd(x[0], c + 0, m, s, idx);
        online_upd(x[1], c + 1, m, s, idx);
        online_upd(x[2], c + 2, m, s, idx);
        online_upd(x[3], c + 3, m, s, idx);
    }
    for (int c = tail0 + tid; c < C; c += 256)
        online_upd(__builtin_nontemporal_load(rp + c), c, m, s, idx);

    sm[tid] = m; ss[tid] = s; si[tid] = idx;
    __syncthreads();
    for (int off = 128; off > 0; off >>= 1) {
        if (tid < off) {
            online_merge(m, s, idx, sm[tid + off], ss[tid + off], si[tid + off]);
            sm[tid] = m; ss[tid] = s; si[tid] = idx;
        }
        __syncthreads();
    }

    if (tid == 0) {
        const int tgt = target[row];
        const float lt = rp[tgt];
        const float logs = logf(s);
        // log_softmax max = -log(s); p = 1/s; ce = -(x_t - lse)
        float p = 1.0f / s;
        p = fminf(fmaxf(p, EPSF), 1.0f - EPSF);
        const float matched = (idx == tgt) ? 1.0f : 0.0f;
        probs[row] = p;
        vf[row]    = (1.0f - matched) * p;        // probs on false set
        vc[row]    = matched * (1.0f - p);        // (1-probs) on correct set
        ce[row]    = -(lt - m - logs);
        mc[row]    = matched;
    }
}

// ---------------- Kernel 2: pairwise kernel sums via WMMA ----------------
// One wave (32 lanes) owns a 16-row block I; loops over all 16-col tiles J.
// For each tile: B[k][n] = exp(-2.5*|p_I[4c+k] - p_J[n]|)  (k=0..3, chunk c),
// A row 0 = vf_I weights, A row 1 = vc_I weights (rows 2..15 zero).
// After 4 chained WMMAs: D row0 = t_f[n] = sum_m vf[m]K[m][n],
//                        D row1 = t_c[n] = sum_m vc[m]K[m][n].
// Layouts per ISA 7.12.2: A(16x4): VGPR0 = K0|K2 (lane halves), VGPR1 = K1|K3;
// B(4x16): same row-striping; D(16x16): VGPRv = row v | row v+8.
__global__ __launch_bounds__(256) void mmce_pairs(
    const float* __restrict__ probs, const float* __restrict__ vf,
    const float* __restrict__ vc, int N, int nW,
    float* __restrict__ pf, float* __restrict__ pc, float* __restrict__ pm)
{
    const int tid  = threadIdx.x;
    const int lane = tid & 31;
    const int gw   = (int)((blockIdx.x * blockDim.x + tid) >> 5);
    if (gw >= nW) return;                 // wave-uniform; EXEC stays all-1s below

    const int i0  = gw << 4;
    const int l16 = lane & 15;

    const float pi  = probs[i0 + l16];    // lane r (and r+16) holds row r of block I
    const float vfi = vf[i0 + l16];
    const float vci = vc[i0 + l16];
    const int kbase = (lane < 16) ? 0 : 2;

    float accf = 0.0f, accc = 0.0f, accm = 0.0f;

    for (int j0 = 0; j0 < N; j0 += 16) {
        const float pj  = probs[j0 + l16];
        const float vfj = vf[j0 + l16];
        const float vcj = vc[j0 + l16];

        v8f d = {0.f, 0.f, 0.f, 0.f, 0.f, 0.f, 0.f, 0.f};
#pragma unroll
        for (int c = 0; c < 4; ++c) {
            const int m0 = 4 * c + kbase;       // source row for this half's VGPR0 slot
            const int m1 = m0 + 1;              // source row for VGPR1 slot
            const float pim0 = __shfl(pi, m0, 32);
            const float pim1 = __shfl(pi, m1, 32);
            v2f b;
            b[0] = expf(-2.5f * fabsf(pim0 - pj));
            b[1] = expf(-2.5f * fabsf(pim1 - pj));

            const float vfm0 = __shfl(vfi, m0, 32);
            const float vcm0 = __shfl(vci, m0, 32);
            const float vfm1 = __shfl(vfi, m1, 32);
            const float vcm1 = __shfl(vci, m1, 32);
            v2f a;
            a[0] = (l16 == 0) ? vfm0 : ((l16 == 1) ? vcm0 : 0.0f);
            a[1] = (l16 == 0) ? vfm1 : ((l16 == 1) ? vcm1 : 0.0f);

            d = __builtin_amdgcn_wmma_f32_16x16x4_f32(
                    false, a, false, b, (short)0, d, false, false);
        }
        // d[0]: row0 (lanes 0-15) = t_f[n], row8 (lanes 16-31) = 0
        // d[1]: row1 (lanes 0-15) = t_c[n], row9 (lanes 16-31) = 0
        accf += d[0] * vfj;
        accc += d[1] * vcj;
        accm += d[1] * vfj;
    }

    // wave32 reduction
#pragma unroll
    for (int off = 16; off > 0; off >>= 1) {
        accf += __shfl_xor(accf, off, 32);
        accc += __shfl_xor(accc, off, 32);
        accm += __shfl_xor(accm, off, 32);
    }
    if (lane == 0) { pf[gw] = accf; pc[gw] = accc; pm[gw] = accm; }
}

// ---------------- Kernel 3: deterministic finalize ----------------
__global__ __launch_bounds__(256) void mmce_finalize(
    const float* __restrict__ ce, const float* __restrict__ mc,
    const float* __restrict__ pf, const float* __restrict__ pc,
    const float* __restrict__ pm, int N, int nW, float* __restrict__ out)
{
    __shared__ float r0[256], r1[256], r2[256], r3[256], r4[256];
    const int tid = threadIdx.x;

    float ce_s = 0.0f, mc_s = 0.0f, f_s = 0.0f, c_s = 0.0f, m_s = 0.0f;
    for (int i = tid; i < N; i += 256) { ce_s += ce[i]; mc_s += mc[i]; }
    for (int i = tid; i < nW; i += 256) { f_s += pf[i]; c_s += pc[i]; m_s += pm[i]; }

    r0[tid] = ce_s; r1[tid] = mc_s; r2[tid] = f_s; r3[tid] = c_s; r4[tid] = m_s;
    __syncthreads();
    for (int off = 128; off > 0; off >>= 1) {
        if (tid < off) {
            r0[tid] += r0[tid + off]; r1[tid] += r1[tid + off];
            r2[tid] += r2[tid + off]; r3[tid] += r3[tid + off];
            r4[tid] += r4[tid + off];
        }
        __syncthreads();
    }

    if (tid == 0) {
        const float n_c = r1[0];
        const float n_f = (float)N - n_c;
        const float s_false = r2[0], s_correct = r3[0], s_mixed = r4[0];
        const float den_f = n_f * n_f;
        const float den_c = n_c * n_c;
        const float den_m = n_f * n_c;
        // safe_div(num, den) = den>0 ? num/max(den,1) : 0   (exactly as reference)
        const float part_false   = (den_f > 0.0f) ? s_false   / fmaxf(den_f, 1.0f) : 0.0f;
        const float part_correct = (den_c > 0.0f) ? s_correct / fmaxf(den_c, 1.0f) : 0.0f;
        const float part_mixed   = 2.0f * ((den_m > 0.0f) ? s_mixed / fmaxf(den_m, 1.0f) : 0.0f);
        const float mmce = sqrtf(part_false + part_correct - part_mixed);
        out[0] = mmce + r0[0] / (float)N;
    }
}

extern "C" void kernel_launch(void* const* d_in, const int* in_sizes, int n_in,
                              void* d_out, int out_size, void* d_ws, size_t ws_size,
                              hipStream_t stream) {
    const float* input  = (const float*)d_in[0];
    const int*   target = (const int*)d_in[1];
    const int N = in_sizes[1];
    const int C = in_sizes[0] / N;
    float* out = (float*)d_out;

    float* ws    = (float*)d_ws;
    float* probs = ws;
    float* vf    = ws + (size_t)N;
    float* vc    = ws + (size_t)2 * N;
    float* ce    = ws + (size_t)3 * N;
    float* mc    = ws + (size_t)4 * N;
    const int nW = N / 16;                 // 16-row blocks (N=4096 -> 256 waves)
    float* pf    = ws + (size_t)5 * N;
    float* pc    = pf + nW;
    float* pm    = pc + nW;

    mmce_rowstats<<<N, 256, 0, stream>>>(input, target, C, probs, vf, vc, ce, mc);

    const int blocks2 = (nW + 7) / 8;      // 8 waves per 256-thread block
    mmce_pairs<<<blocks2, 256, 0, stream>>>(probs, vf, vc, N, nW, pf, pc, pm);

    mmce_finalize<<<1, 256, 0, stream>>>(ce, mc, pf, pc, pm, N, nW, out);
}